// RiemannianBatchNorm_40372692583292
// MI455X (gfx1250) — compile-verified
//
#include <hip/hip_runtime.h>
#include <math.h>

// ---------------------------------------------------------------------------
// Riemannian SPD batch-norm, matmul-only formulation for CDNA5 (gfx1250).
// All matrix functions (sqrtm/invsqrtm/logm/expm/pow) are computed with
// Newton-Schulz / scaling-and-squaring iterations built on a 64x64 f32 GEMM
// that runs on V_WMMA_F32_16X16X4_F32 (wave32, 16 waves per block, one 16x16
// output tile per wave).  Fragments for the whole K dimension are loaded
// before the WMMA chain so the 16 matrix ops issue back-to-back behind a
// single s_wait_dscnt instead of one wait per step.
// ---------------------------------------------------------------------------

typedef __attribute__((ext_vector_type(2))) float v2f;
typedef __attribute__((ext_vector_type(8))) float v8f;

#define NT 512          // 16 wave32 per block
#define NMAT 4096       // 64*64
#define EPW 8           // NMAT / NT elements per thread
#define NS_ITERS 12     // Newton-Schulz iterations for sqrt/invsqrt
#define LOG_SQRTS 3     // inverse scaling-and-squaring depth for logm
#define LOG_TERMS 8     // Mercator series terms for log(I+E)
#define EXP_TERMS 8     // Taylor terms for expm
#define EXP_SQUARINGS 5 // expm scaling 2^-5

#define BATCH 8192
#define NPART 64        // partial-sum blocks for batch reductions
#define PER_PART (BATCH / NPART)

// ------------------------- 64x64 GEMM on WMMA f32 --------------------------
// C = A * B, all pointers into LDS (or generic). Caller must __syncthreads()
// after any writes to A/B before calling, and after the call before reading C.
// Requires all 512 threads active (EXEC all ones for WMMA).
__device__ __forceinline__ void mm64(float* C, const float* A, const float* B,
                                     int tid) {
  const int wave = tid >> 5;
  const int lane = tid & 31;
  const int tm = (wave >> 2) << 4;  // tile row base
  const int tn = (wave & 3) << 4;   // tile col base
  const int l15 = lane & 15;
  const int hi = lane >> 4;         // lanes 16..31 hold K+2 / M+8 halves

  const float* Arow = A + (tm + l15) * 64 + (hi << 1);   // A[m][kbase + 2*hi]
  const float* Bcol = B + ((hi << 1) * 64) + tn + l15;   // B[kbase + 2*hi][n]

  // Load all fragments first so the WMMA chain is not serialized on LDS.
  v2f a[16], b[16];
#pragma unroll
  for (int k = 0; k < 16; ++k) {
    a[k].x = Arow[4 * k];
    a[k].y = Arow[4 * k + 1];
    b[k].x = Bcol[(4 * k) * 64];
    b[k].y = Bcol[(4 * k + 1) * 64];
  }

  v8f acc = {0.f, 0.f, 0.f, 0.f, 0.f, 0.f, 0.f, 0.f};
#pragma unroll
  for (int k = 0; k < 16; ++k) {
    acc = __builtin_amdgcn_wmma_f32_16x16x4_f32(
        /*neg_a=*/false, a[k], /*neg_b=*/false, b[k],
        /*c_mod=*/(short)0, acc, /*reuse_a=*/false, /*reuse_b=*/false);
  }

  // D layout: VGPR r -> M = tm + r + 8*hi, N = tn + l15
  float* Crow = C + (tm + (hi << 3)) * 64 + tn + l15;
#pragma unroll
  for (int r = 0; r < 8; ++r) Crow[r * 64] = acc[r];
}

// ------------------------- block-wide scalar reduce ------------------------
__device__ __forceinline__ float block_sum(float v, float* red, int tid) {
  red[tid] = v;
  __syncthreads();
#pragma unroll
  for (int s = NT / 2; s > 0; s >>= 1) {
    if (tid < s) red[tid] += red[tid + s];
    __syncthreads();
  }
  float r = red[0];
  __syncthreads();
  return r;
}

// ---------------- coupled Newton-Schulz sqrt / invsqrt ---------------------
// Y ~= sqrt(A), Z ~= invsqrt(A).  A (any address space) is preserved and must
// be distinct from the four LDS pool buffers p0..p3.  Results returned as
// pointers into the pool (ping-pong, no copies).
__device__ void ns_sqrt_dev(const float* A, float*& Yo, float*& Zo, float* p0,
                            float* p1, float* p2, float* p3, float* red,
                            int tid) {
  float* Y = p0;
  float* Z = p1;
  float* T = p2;
  float* SP = p3;

  float acc = 0.f;
#pragma unroll
  for (int i = 0; i < EPW; ++i) {
    float x = A[(i << 9) | tid];
    acc += x * x;
  }
  float nrm = sqrtf(block_sum(acc, red, tid));
  float inv = (nrm > 0.f) ? 1.0f / nrm : 0.f;

#pragma unroll
  for (int i = 0; i < EPW; ++i) {
    int e = (i << 9) | tid;
    int r = e >> 6, c = e & 63;
    Y[e] = A[e] * inv;
    Z[e] = (r == c) ? 1.0f : 0.0f;
  }
  __syncthreads();

  for (int it = 0; it < NS_ITERS; ++it) {
    mm64(T, Z, Y, tid);
    __syncthreads();
#pragma unroll
    for (int i = 0; i < EPW; ++i) {
      int e = (i << 9) | tid;
      int r = e >> 6, c = e & 63;
      T[e] = ((r == c) ? 1.5f : 0.0f) - 0.5f * T[e];  // T = (3I - Z Y)/2
    }
    __syncthreads();
    mm64(SP, Y, T, tid);  // Y' = Y T
    __syncthreads();
    { float* t = Y; Y = SP; SP = t; }
    mm64(SP, T, Z, tid);  // Z' = T Z
    __syncthreads();
    { float* t = Z; Z = SP; SP = t; }
  }

  float sq = sqrtf(nrm);
  float isq = (sq > 0.f) ? 1.0f / sq : 0.f;
#pragma unroll
  for (int i = 0; i < EPW; ++i) {
    int e = (i << 9) | tid;
    Y[e] *= sq;
    Z[e] *= isq;
  }
  __syncthreads();
  Yo = Y;
  Zo = Z;
}

// ---------------------------- logm (SPD) -----------------------------------
// Inverse scaling-and-squaring: W -> W^(1/2^k) via NS roots, then Mercator
// series on E = W^(1/2^k) - I, result scaled by 2^k.  Returns pointer to the
// result (one of the 5 distinct LDS buffers {W,q0..q3}).
__device__ float* logm_dev(float* W, float* q0, float* q1, float* q2,
                           float* q3, float* red, int tid) {
  for (int j = 0; j < LOG_SQRTS; ++j) {
    float *Y, *Z;
    ns_sqrt_dev(W, Y, Z, q0, q1, q2, q3, red, tid);
    float* all[5] = {W, q0, q1, q2, q3};
    W = Y;  // W <- sqrt(W) by pointer swap
    float* np[4];
    int k = 0;
    for (int i2 = 0; i2 < 5; ++i2)
      if (all[i2] != W) np[k++] = all[i2];
    q0 = np[0]; q1 = np[1]; q2 = np[2]; q3 = np[3];
  }

  // E = W - I (q0),  S = E (q1)
#pragma unroll
  for (int i = 0; i < EPW; ++i) {
    int e = (i << 9) | tid;
    int r = e >> 6, c = e & 63;
    float v = W[e] - ((r == c) ? 1.0f : 0.0f);
    q0[e] = v;
    q1[e] = v;
  }
  __syncthreads();

  float* P = q0;   // current power E^(i-1)
  float* Pa = q2;
  float* Pb = q3;
  for (int i = 2; i <= LOG_TERMS; ++i) {
    mm64(Pa, P, q0, tid);
    __syncthreads();
    P = Pa;
    { float* t = Pa; Pa = Pb; Pb = t; }
    float coef = ((i & 1) ? 1.0f : -1.0f) / (float)i;  // (-1)^(i+1)/i
#pragma unroll
    for (int k2 = 0; k2 < EPW; ++k2) {
      int e = (k2 << 9) | tid;
      q1[e] += coef * P[e];
    }
    __syncthreads();
  }

  const float scl = (float)(1 << LOG_SQRTS);
#pragma unroll
  for (int i = 0; i < EPW; ++i) q1[(i << 9) | tid] *= scl;
  __syncthreads();
  return q1;
}

// ---------------------------- expm -----------------------------------------
// R = exp(M * pref) via scaling-and-squaring.  M (in LDS buffer E) is scaled
// in place; R ping-pongs between b1/b2; returns pointer to the result.
__device__ float* expm_dev(float* E, float pref, float* b1, float* b2,
                           int tid) {
  const float s = pref * (1.0f / (float)(1 << EXP_SQUARINGS));
  float* R = b1;
  float* T = b2;
#pragma unroll
  for (int i = 0; i < EPW; ++i) {
    int e = (i << 9) | tid;
    int r = e >> 6, c = e & 63;
    E[e] *= s;
    R[e] = (r == c) ? 1.0f : 0.0f;
  }
  __syncthreads();

  for (int i = EXP_TERMS; i >= 1; --i) {
    mm64(T, E, R, tid);
    __syncthreads();
    float inv = 1.0f / (float)i;
#pragma unroll
    for (int k = 0; k < EPW; ++k) {
      int e = (k << 9) | tid;
      int r = e >> 6, c = e & 63;
      T[e] = ((r == c) ? 1.0f : 0.0f) + T[e] * inv;  // R' = I + (1/i) E R
    }
    __syncthreads();
    { float* t = R; R = T; T = t; }
  }
  for (int sq = 0; sq < EXP_SQUARINGS; ++sq) {
    mm64(T, R, R, tid);
    __syncthreads();
    { float* t = R; R = T; T = t; }
  }
  return R;
}

// =========================== kernels =======================================

// partial[blk][e] = sum over PER_PART matrices of src chunk (deterministic)
__global__ void batch_sum_partial(const float* __restrict__ src,
                                  float* __restrict__ partial) {
  const int blk = blockIdx.x;
  const float* base = src + (size_t)blk * PER_PART * NMAT;
  for (int e = threadIdx.x; e < NMAT; e += blockDim.x) {
    float s = 0.f;
    for (int b = 0; b < PER_PART; ++b) s += base[(size_t)b * NMAT + e];
    partial[(size_t)blk * NMAT + e] = s;
  }
}

__global__ void batch_sum_final(const float* __restrict__ partial,
                                float* __restrict__ dst, float scale) {
  int e = blockIdx.x * blockDim.x + threadIdx.x;
  if (e < NMAT) {
    float s = 0.f;
    for (int p = 0; p < NPART; ++p) s += partial[(size_t)p * NMAT + e];
    dst[e] = s * scale;
  }
}

// S = sqrt(A), SI = invsqrt(A); one block of 512 threads.
__global__ void sqrt_invsqrt_kernel(const float* __restrict__ A,
                                    float* __restrict__ Sout,
                                    float* __restrict__ SIout) {
  extern __shared__ float smem[];
  float* b0 = smem;
  float* b1 = smem + NMAT;
  float* b2 = smem + 2 * NMAT;
  float* b3 = smem + 3 * NMAT;
  float* red = smem + 4 * NMAT;
  const int tid = threadIdx.x;
  float *Y, *Z;
  ns_sqrt_dev(A, Y, Z, b0, b1, b2, b3, red, tid);
#pragma unroll
  for (int i = 0; i < EPW; ++i) {
    int e = (i << 9) | tid;
    Sout[e] = Y[e];
    SIout[e] = Z[e];
  }
}

// G[b] = logm(si x_b si); varpart[b] = ||G_b||_F^2.  One block per sample.
__global__ void log_whiten_kernel(const float* __restrict__ x,
                                  const float* __restrict__ si,
                                  float* __restrict__ G,
                                  float* __restrict__ varpart) {
  extern __shared__ float smem[];
  float* b0 = smem;
  float* b1 = smem + NMAT;
  float* b2 = smem + 2 * NMAT;
  float* b3 = smem + 3 * NMAT;
  float* b4 = smem + 4 * NMAT;
  float* red = smem + 5 * NMAT;
  const int tid = threadIdx.x;
  const int b = blockIdx.x;
  const float* xb = x + (size_t)b * NMAT;

#pragma unroll
  for (int i = 0; i < EPW; ++i) {
    int e = (i << 9) | tid;
    b0[e] = xb[e];   // X
    b1[e] = si[e];   // SI
  }
  __syncthreads();
  mm64(b2, b1, b0, tid);  // SI @ X
  __syncthreads();
  mm64(b3, b2, b1, tid);  // W = SI @ X @ SI
  __syncthreads();

  float* L = logm_dev(b3, b0, b1, b2, b4, red, tid);

  float acc = 0.f;
#pragma unroll
  for (int i = 0; i < EPW; ++i) {
    float v = L[(i << 9) | tid];
    acc += v * v;
  }
  float f2 = block_sum(acc, red, tid);
  if (tid == 0) varpart[b] = f2;

  float* Gb = G + (size_t)b * NMAT;
#pragma unroll
  for (int i = 0; i < EPW; ++i) {
    int e = (i << 9) | tid;
    Gb[e] = L[e];
  }
}

// mean1 = s0 @ expm(t) @ s0; one block.
__global__ void karcher_update_kernel(const float* __restrict__ t,
                                      const float* __restrict__ s0,
                                      float* __restrict__ mean1) {
  extern __shared__ float smem[];
  float* b0 = smem;
  float* b1 = smem + NMAT;
  float* b2 = smem + 2 * NMAT;
  float* b3 = smem + 3 * NMAT;
  const int tid = threadIdx.x;

#pragma unroll
  for (int i = 0; i < EPW; ++i) {
    int e = (i << 9) | tid;
    b0[e] = t[e];
  }
  __syncthreads();
  float* R = expm_dev(b0, 1.0f, b1, b2, tid);  // exp(t)
  float* T = (R == b1) ? b2 : b1;
#pragma unroll
  for (int i = 0; i < EPW; ++i) {
    int e = (i << 9) | tid;
    b3[e] = s0[e];
  }
  __syncthreads();
  mm64(T, b3, R, tid);  // s0 @ exp(t)
  __syncthreads();
  mm64(R, T, b3, tid);  // (s0 exp(t)) @ s0
  __syncthreads();
#pragma unroll
  for (int i = 0; i < EPW; ++i) {
    int e = (i << 9) | tid;
    mean1[e] = R[e];
  }
}

// p = scale / (sqrt(mean(varpart)) + eps)
__global__ void var_p_kernel(const float* __restrict__ varpart,
                             const float* __restrict__ scale,
                             float* __restrict__ pOut) {
  __shared__ float red[256];
  const int tid = threadIdx.x;
  float s = 0.f;
  for (int i = tid; i < BATCH; i += 256) s += varpart[i];
  red[tid] = s;
  __syncthreads();
#pragma unroll
  for (int t = 128; t > 0; t >>= 1) {
    if (tid < t) red[tid] += red[tid + t];
    __syncthreads();
  }
  if (tid == 0) {
    float var = red[0] / (float)BATCH;
    float sd = sqrtf(var);
    pOut[0] = scale[0] / (sd + 1e-5f);
  }
}

// out_b = ss @ exp(p * G_b) @ ss, in place over G (= d_out). Block per sample.
__global__ void power_shift_kernel(float* __restrict__ G,
                                   const float* __restrict__ pPtr,
                                   const float* __restrict__ ss) {
  extern __shared__ float smem[];
  float* b0 = smem;
  float* b1 = smem + NMAT;
  float* b2 = smem + 2 * NMAT;
  float* b3 = smem + 3 * NMAT;
  const int tid = threadIdx.x;
  const int b = blockIdx.x;
  float* Gb = G + (size_t)b * NMAT;
  const float p = pPtr[0];

#pragma unroll
  for (int i = 0; i < EPW; ++i) {
    int e = (i << 9) | tid;
    b0[e] = Gb[e];
  }
  __syncthreads();
  float* R = expm_dev(b0, p, b1, b2, tid);  // exp(p G) = (si x si)^p
  float* T = (R == b1) ? b2 : b1;
#pragma unroll
  for (int i = 0; i < EPW; ++i) {
    int e = (i << 9) | tid;
    b3[e] = ss[e];
  }
  __syncthreads();
  mm64(T, b3, R, tid);  // ss @ Y
  __syncthreads();
  mm64(R, T, b3, tid);  // (ss Y) @ ss
  __syncthreads();
#pragma unroll
  for (int i = 0; i < EPW; ++i) {
    int e = (i << 9) | tid;
    Gb[e] = R[e];
  }
}

// =========================== host launcher =================================

extern "C" void kernel_launch(void* const* d_in, const int* in_sizes, int n_in,
                              void* d_out, int out_size, void* d_ws,
                              size_t ws_size, hipStream_t stream) {
  const float* x = (const float*)d_in[0];      // [8192, 64, 64]
  const float* shift = (const float*)d_in[1];  // [64, 64]
  const float* scale = (const float*)d_in[2];  // [1]
  float* out = (float*)d_out;                  // [8192, 64, 64]
  float* ws = (float*)d_ws;

  // ws layout (floats)
  float* ws_partial = ws;                         // 64*4096
  float* ws_mean0 = ws + (size_t)NPART * NMAT;    // 4096
  float* ws_s0 = ws_mean0 + NMAT;
  float* ws_si0 = ws_s0 + NMAT;
  float* ws_t = ws_si0 + NMAT;
  float* ws_mean1 = ws_t + NMAT;
  float* ws_s1 = ws_mean1 + NMAT;
  float* ws_si1 = ws_s1 + NMAT;
  float* ws_ss = ws_si1 + NMAT;
  float* ws_ssi = ws_ss + NMAT;
  float* ws_varpart = ws_ssi + NMAT;              // 8192
  float* ws_p = ws_varpart + BATCH;               // 1

  const size_t SM_NS = (size_t)(4 * NMAT + NT) * sizeof(float);
  const size_t SM_LOG = (size_t)(5 * NMAT + NT) * sizeof(float);
  const size_t SM_EXP = (size_t)(4 * NMAT + NT) * sizeof(float);
  const float invB = 1.0f / (float)BATCH;

  // mean0 = arithmetic mean of x
  batch_sum_partial<<<NPART, 256, 0, stream>>>(x, ws_partial);
  batch_sum_final<<<NMAT / 256, 256, 0, stream>>>(ws_partial, ws_mean0, invB);

  // s0 / si0 of mean0
  sqrt_invsqrt_kernel<<<1, NT, SM_NS, stream>>>(ws_mean0, ws_s0, ws_si0);

  // Karcher tangent mean: t = mean_b logm(si0 x_b si0)   (stage logs in d_out)
  log_whiten_kernel<<<BATCH, NT, SM_LOG, stream>>>(x, ws_si0, out, ws_varpart);
  batch_sum_partial<<<NPART, 256, 0, stream>>>(out, ws_partial);
  batch_sum_final<<<NMAT / 256, 256, 0, stream>>>(ws_partial, ws_t, invB);

  // mean1 = s0 expm(t) s0 ; si1 = mean1^{-1/2}
  karcher_update_kernel<<<1, NT, SM_EXP, stream>>>(ws_t, ws_s0, ws_mean1);
  sqrt_invsqrt_kernel<<<1, NT, SM_NS, stream>>>(ws_mean1, ws_s1, ws_si1);

  // G_b = logm(si1 x_b si1) -> d_out ; per-sample ||G||_F^2 -> varpart
  log_whiten_kernel<<<BATCH, NT, SM_LOG, stream>>>(x, ws_si1, out, ws_varpart);

  // p = scale / (std + eps)
  var_p_kernel<<<1, 256, 0, stream>>>(ws_varpart, scale, ws_p);

  // ss = shift^{1/2}
  sqrt_invsqrt_kernel<<<1, NT, SM_NS, stream>>>(shift, ws_ss, ws_ssi);

  // out_b = ss exp(p G_b) ss (in place over d_out)
  power_shift_kernel<<<BATCH, NT, SM_EXP, stream>>>(out, ws_p, ws_ss);
}